// DecisionHead_56779467653346
// MI455X (gfx1250) — compile-verified
//
#include <hip/hip_runtime.h>
#include <hip/hip_bf16.h>

typedef float v2f __attribute__((ext_vector_type(2)));
typedef float v8f __attribute__((ext_vector_type(8)));

#define C_IN 1024
#define NB   256
#define NA   32
#define HW   196
#define C_OUT 1024

// ---------------------------------------------------------------------------
// Kernel 1: pooled[b,c] = mean(relu(x[b,c,:,:]))  -- bandwidth bound (205 MB)
// One wave32 per (b,c) row of 196 floats (= 49 float4, 16B-aligned rows).
// ---------------------------------------------------------------------------
__global__ __launch_bounds__(256) void k_pool_relu(const float* __restrict__ x,
                                                   float* __restrict__ pooled) {
    const int lane = threadIdx.x & 31;
    const int wave = threadIdx.x >> 5;
    const long row = (long)blockIdx.x * 8 + wave;     // 0 .. B*C_in-1
    const float4* v = (const float4*)(x + row * HW);

    float4 p = v[lane];                                // float4 0..31
    float s = fmaxf(p.x, 0.f) + fmaxf(p.y, 0.f) + fmaxf(p.z, 0.f) + fmaxf(p.w, 0.f);
    if (lane < 17) {                                   // float4 32..48
        float4 q = v[32 + lane];
        s += fmaxf(q.x, 0.f) + fmaxf(q.y, 0.f) + fmaxf(q.z, 0.f) + fmaxf(q.w, 0.f);
    }
    #pragma unroll
    for (int o = 16; o > 0; o >>= 1) s += __shfl_xor(s, o, 32);
    if (lane == 0) pooled[row] = s * (1.0f / (float)HW);
}

// ---------------------------------------------------------------------------
// Kernel 2: logits[256,32] = pooled[256,1024] x fc1_weight[32,1024]^T
// Native fp32 WMMA: V_WMMA_F32_16X16X4_F32.  32 tiles of 16x16, one per wave.
// A frag (16x4 f32): lane&15 = M, (lane>>4) selects K={v+0,v+1} vs {v+2,v+3}.
// B frag mirrors per K-row; C/D: VGPR v, half g -> M = v + 8g, N = lane&15.
// ---------------------------------------------------------------------------
__global__ __launch_bounds__(256) void k_gemm_logits(const float* __restrict__ pooled,
                                                     const float* __restrict__ w,
                                                     float* __restrict__ logits) {
    const int lane  = threadIdx.x & 31;
    const int wave  = threadIdx.x >> 5;
    const int tile  = blockIdx.x * 8 + wave;      // 0..31
    const int tileM = (tile & 15) * 16;           // 16 M-tiles
    const int tileN = (tile >> 4) * 16;           // 2  N-tiles
    const int half  = lane >> 4;                  // 0 | 1
    const int l15   = lane & 15;

    const float* aPtr = pooled + (tileM + l15) * C_IN + half * 2;  // A row M=l15
    const float* bPtr = w      + (tileN + l15) * C_IN + half * 2;  // B col N=l15 (w is [A,C_in])

    v8f acc = {};
    for (int k = 0; k < C_IN; k += 4) {
        v2f a, b;
        a.x = aPtr[k]; a.y = aPtr[k + 1];
        b.x = bPtr[k]; b.y = bPtr[k + 1];
        acc = __builtin_amdgcn_wmma_f32_16x16x4_f32(
            /*neg_a=*/false, a, /*neg_b=*/false, b,
            /*c_mod=*/(short)0, acc, /*reuse_a=*/false, /*reuse_b=*/false);
    }
    #pragma unroll
    for (int v = 0; v < 8; ++v) {
        const int m = tileM + v + 8 * half;
        const int n = tileN + l15;
        logits[m * NA + n] = acc[v];
    }
}

// ---------------------------------------------------------------------------
// Kernel 3: per-row log-softmax + Gumbel noise + argmax (wave32 per row).
// softmax is monotone, so argmax((logp+g)/T) == argmax of the perturbed logit.
// Tie-break: lowest index (matches jnp.argmax).
// ---------------------------------------------------------------------------
__global__ __launch_bounds__(256) void k_gumbel_argmax(const float* __restrict__ logits,
                                                       const float* __restrict__ gu,
                                                       int* __restrict__ sampled,
                                                       float* __restrict__ out_sa) {
    const int lane = threadIdx.x & 31;
    const int b    = blockIdx.x * 8 + (threadIdx.x >> 5);   // one wave per row

    const float l = logits[b * NA + lane];
    float m = l;
    #pragma unroll
    for (int o = 16; o > 0; o >>= 1) m = fmaxf(m, __shfl_xor(m, o, 32));
    float e = __expf(l - m);
    float s = e;
    #pragma unroll
    for (int o = 16; o > 0; o >>= 1) s += __shfl_xor(s, o, 32);
    const float logp = l - m - __logf(s);

    const float u = gu[b * NA + lane];
    const float g = -__logf(-__logf(u + 1e-10f) + 1e-10f);
    float y  = (logp + g) * 2.0f;                           // /TEMPERATURE(0.5)
    int   idx = lane;
    #pragma unroll
    for (int o = 16; o > 0; o >>= 1) {
        const float oy = __shfl_xor(y, o, 32);
        const int   oi = __shfl_xor(idx, o, 32);
        if (oy > y || (oy == y && oi < idx)) { y = oy; idx = oi; }
    }
    if (lane == 0) { sampled[b] = idx; out_sa[b] = (float)idx; }
}

// ---------------------------------------------------------------------------
// Kernel 4: selected_channels[b,:] = channel_gates[sampled[b],:]  (row gather)
// substitute == one-hot(argmax) in forward value, so this is exact.
// ---------------------------------------------------------------------------
__global__ __launch_bounds__(256) void k_gather(const int* __restrict__ sampled,
                                                const float* __restrict__ gates,
                                                float* __restrict__ out) {
    const int b  = blockIdx.x;
    const int sa = sampled[b];
    const float4* src = (const float4*)(gates + (long)sa * C_OUT);
    float4*       dst = (float4*)(out + (long)b * C_OUT);
    dst[threadIdx.x] = src[threadIdx.x];                    // 256 x float4 = 1024 f32
}

extern "C" void kernel_launch(void* const* d_in, const int* in_sizes, int n_in,
                              void* d_out, int out_size, void* d_ws, size_t ws_size,
                              hipStream_t stream) {
    const float* x     = (const float*)d_in[0];   // [256,1024,14,14]
    const float* gu    = (const float*)d_in[1];   // [256,32]
    const float* fc1_w = (const float*)d_in[2];   // [32,1024]
    const float* gates = (const float*)d_in[3];   // [32,1024]
    float* out = (float*)d_out;                   // 256 + 256*1024 floats

    char* ws = (char*)d_ws;
    float* pooled  = (float*)(ws);                           // 256*1024 f32 = 1 MB
    float* logits  = (float*)(ws + (size_t)NB * C_IN * 4);   // 256*32 f32
    int*   sampled = (int*)  (ws + (size_t)NB * C_IN * 4 + (size_t)NB * NA * 4);

    float* out_sa  = out;            // first 256 floats: sampled_actions
    float* out_sel = out + NB;       // then [256,1024] selected_channels

    // 1) relu+pool: 32768 blocks x 8 waves, one wave per (b,c) row
    k_pool_relu<<<(NB * C_IN) / 8, 256, 0, stream>>>(x, pooled);
    // 2) fp32 WMMA GEMM: 32 tiles = 4 blocks x 8 waves
    k_gemm_logits<<<4, 256, 0, stream>>>(pooled, fc1_w, logits);
    // 3) gumbel softmax argmax: 256 rows, wave per row
    k_gumbel_argmax<<<NB / 8, 256, 0, stream>>>(logits, gu, sampled, out_sa);
    // 4) gather gate rows
    k_gather<<<NB, 256, 0, stream>>>(sampled, gates, out_sel);
}